// CustomMSDeformableAttention_14465449853376
// MI455X (gfx1250) — compile-verified
//
#include <hip/hip_runtime.h>
#include <hip/hip_bf16.h>

#define EMBED    256
#define HEADS    8
#define POINTS   4
#define HEAD_DIM 32
#define SH       200
#define SW       200
#define BS       2
#define NQ       40000
#define NV       (SH * SW)

typedef __attribute__((ext_vector_type(2))) float v2f;
typedef __attribute__((ext_vector_type(8))) float v8f;

// ---------------------------------------------------------------------------
// Repack W[K=256][N] (row-major) into WMMA B-fragment order:
// for tile (kt, nt): 32 lanes x 2 floats, lane l holds
//   b.x = W[kt*4 + (l>=16?2:0) + 0][nt*16 + (l&15)]
//   b.y = W[kt*4 + (l>=16?2:0) + 1][nt*16 + (l&15)]
// so the GEMM inner loop does one coalesced float2 load per B fragment.
// ---------------------------------------------------------------------------
__global__ void pack_w_kernel(const float* __restrict__ W, float* __restrict__ Wp, int N) {
    int t = blockIdx.x * blockDim.x + threadIdx.x;
    int total = 256 * N;
    if (t >= total) return;
    int tile   = t >> 6;
    int within = t & 63;
    int lane   = within >> 1;
    int kk     = within & 1;
    int NT     = N >> 4;
    int kt     = tile / NT;
    int nt     = tile - kt * NT;
    int k      = kt * 4 + ((lane >= 16) ? 2 : 0) + kk;
    int n      = nt * 16 + (lane & 15);
    Wp[t] = W[k * N + n];
}

// ---------------------------------------------------------------------------
// fp32 WMMA GEMM: out[M,Nfull] = A[M,256] @ W[256,Nfull] + bias (+ residual)
// One wave computes 16 rows x (NT*16) cols; blockIdx.y selects the column
// chunk (nt0 = blockIdx.y*NT).  NT<=8 keeps C at <=64 VGPRs -> no spills.
// Inner loop: 1x b64 A load + NT-wide b64 B-load clause + NT WMMAs with
// per-fragment s_wait_loadcnt (verified in disasm).
// ---------------------------------------------------------------------------
template <int NT, bool HAS_BIAS, bool HAS_RES>
__global__ __launch_bounds__(256) void gemm_wmma_kernel(
        const float* __restrict__ A,
        const float* __restrict__ Wp,
        const float* __restrict__ bias,
        const float* __restrict__ residual,
        float* __restrict__ out, int M, int Nfull) {
    const int ntf  = Nfull >> 4;               // total column tiles in Wp
    const int nt0  = blockIdx.y * NT;          // this wave's first column tile
    int wave = (int)((blockIdx.x * (long long)blockDim.x + threadIdx.x) >> 5);
    int lane = threadIdx.x & 31;
    int m0   = wave * 16;
    if (m0 >= M) return;  // wave-uniform: EXEC stays all-ones for WMMA

    v8f c[NT];
#pragma unroll
    for (int t = 0; t < NT; ++t)
        c[t] = (v8f){0.f, 0.f, 0.f, 0.f, 0.f, 0.f, 0.f, 0.f};

    const int hi = lane >> 4;                  // K-half select (0 -> K0/1, 1 -> K2/3)
    const float* aptr = A + (long long)(m0 + (lane & 15)) * 256 + hi * 2;
    const float* wptr = Wp + (long long)nt0 * 64 + lane * 2;

#pragma unroll 1
    for (int k0 = 0; k0 < 256; k0 += 4) {
        v2f a = *(const v2f*)(aptr + k0);
        const float* wp = wptr + (k0 >> 2) * (ntf * 64);
#pragma unroll
        for (int t = 0; t < NT; ++t) {
            v2f b = *(const v2f*)(wp + t * 64);
            // (neg_a, A, neg_b, B, c_mod, C, reuse_a, reuse_b)
            c[t] = __builtin_amdgcn_wmma_f32_16x16x4_f32(
                false, a, false, b, (short)0, c[t], false, false);
        }
    }

    // C/D layout: lanes 0-15 -> rows m0+j, col N0+lane; lanes 16-31 -> rows m0+8+j
    int rbase = m0 + hi * 8;
    int col   = lane & 15;
#pragma unroll
    for (int t = 0; t < NT; ++t) {
        int n = (nt0 + t) * 16 + col;
        float bv = HAS_BIAS ? bias[n] : 0.f;
#pragma unroll
        for (int j = 0; j < 8; ++j) {
            long long r = rbase + j;
            float v = c[t][j] + bv;
            if (HAS_RES) v += residual[r * Nfull + n];
            out[r * Nfull + n] = v;
        }
    }
}

// ---------------------------------------------------------------------------
// Deformable sampling: one wave per (b, q, head); 32 lanes = HEAD_DIM.
// Softmax over the 4 points, bilinear zero-padded gather from projected V.
// Each corner gather: 32 lanes x 4B = one 128B line (V fits in 192MB L2).
// ---------------------------------------------------------------------------
__global__ void msda_sample_kernel(const float* __restrict__ V,
                                   const float* __restrict__ off_ws,
                                   const float* __restrict__ attn_ws,
                                   const float* __restrict__ ref,
                                   float* __restrict__ out_attn) {
    long long gw = ((long long)blockIdx.x * blockDim.x + threadIdx.x) >> 5;
    int lane = threadIdx.x & 31;
    const long long total = (long long)BS * NQ * HEADS;
    if (gw >= total) return;

    int h        = (int)(gw % HEADS);
    long long bq = gw / HEADS;
    int q        = (int)(bq % NQ);
    int b        = (int)(bq / NQ);
    long long row = (long long)b * NQ + q;

    float rx = ref[row * 2 + 0];
    float ry = ref[row * 2 + 1];
    const float* op = off_ws + row * (HEADS * POINTS * 2) + h * (POINTS * 2);
    const float* ap = attn_ws + row * (HEADS * POINTS) + h * POINTS;

    // softmax over POINTS (wave-uniform, computed redundantly per lane)
    float a0 = ap[0], a1 = ap[1], a2 = ap[2], a3 = ap[3];
    float mx = fmaxf(fmaxf(a0, a1), fmaxf(a2, a3));
    float e0 = __expf(a0 - mx), e1 = __expf(a1 - mx);
    float e2 = __expf(a2 - mx), e3 = __expf(a3 - mx);
    float inv = 1.f / (e0 + e1 + e2 + e3);
    float aw[POINTS] = {e0 * inv, e1 * inv, e2 * inv, e3 * inv};

    const float* vb = V + (long long)b * NV * EMBED + h * HEAD_DIM + lane;
    float acc = 0.f;

#pragma unroll
    for (int p = 0; p < POINTS; ++p) {
        // loc = ref + off/(W,H); px = loc_x*W - 0.5 = rx*W + off_x - 0.5
        float px = rx * (float)SW + op[p * 2 + 0] - 0.5f;
        float py = ry * (float)SH + op[p * 2 + 1] - 0.5f;
        float fx0 = floorf(px), fy0 = floorf(py);
        int x0 = (int)fx0, y0 = (int)fy0;
        int x1 = x0 + 1,   y1 = y0 + 1;
        float fx = px - fx0, fy = py - fy0;
        float w = aw[p];
        float w00 = (1.f - fx) * (1.f - fy) * w;
        float w10 = fx * (1.f - fy) * w;
        float w01 = (1.f - fx) * fy * w;
        float w11 = fx * fy * w;
        bool vx0 = (x0 >= 0) & (x0 < SW);
        bool vx1 = (x1 >= 0) & (x1 < SW);
        bool vy0 = (y0 >= 0) & (y0 < SH);
        bool vy1 = (y1 >= 0) & (y1 < SH);
        if (vx0 & vy0) acc += w00 * vb[(long long)(y0 * SW + x0) * EMBED];
        if (vx1 & vy0) acc += w10 * vb[(long long)(y0 * SW + x1) * EMBED];
        if (vx0 & vy1) acc += w01 * vb[(long long)(y1 * SW + x0) * EMBED];
        if (vx1 & vy1) acc += w11 * vb[(long long)(y1 * SW + x1) * EMBED];
    }

    out_attn[row * EMBED + h * HEAD_DIM + lane] = acc;
}

// ---------------------------------------------------------------------------
// Host-side orchestration
// ---------------------------------------------------------------------------
extern "C" void kernel_launch(void* const* d_in, const int* in_sizes, int n_in,
                              void* d_out, int out_size, void* d_ws, size_t ws_size,
                              hipStream_t stream) {
    const float* query   = (const float*)d_in[0];   // [BS, NQ, 256]
    const float* value   = (const float*)d_in[1];   // [BS, NV, 256]
    const float* ref     = (const float*)d_in[2];   // [BS, NQ, 1, 2]
    // d_in[3]: spatial_shapes (int32) — constant [(200,200)], baked in
    const float* W_value = (const float*)d_in[4];
    const float* b_value = (const float*)d_in[5];
    const float* W_off   = (const float*)d_in[6];
    const float* b_off   = (const float*)d_in[7];
    const float* W_attn  = (const float*)d_in[8];
    const float* b_attn  = (const float*)d_in[9];
    const float* W_out   = (const float*)d_in[10];
    const float* b_out   = (const float*)d_in[11];
    float* out = (float*)d_out;

    // Workspace layout (bytes)
    char* ws = (char*)d_ws;
    const size_t OFF_WP_VAL  = 0;                       // 256*256*4 = 262144
    const size_t OFF_WP_OUT  = 262144;                  // 262144
    const size_t OFF_WP_OFF  = 524288;                  // 256*64*4 = 65536
    const size_t OFF_WP_ATT  = 589824;                  // 256*32*4 = 32768
    const size_t OFF_V       = 1 << 20;                 // 80000*256*4 = 81,920,000
    const size_t OFF_OFFWS   = OFF_V + 81920000ull;     // 80000*64*4 = 20,480,000
    const size_t OFF_ATTWS   = OFF_OFFWS + 20480000ull; // 80000*32*4 = 10,240,000
    const size_t OFF_OATTN   = OFF_ATTWS + 10240000ull; // 81,920,000

    float* wp_value = (float*)(ws + OFF_WP_VAL);
    float* wp_out   = (float*)(ws + OFF_WP_OUT);
    float* wp_off   = (float*)(ws + OFF_WP_OFF);
    float* wp_attn  = (float*)(ws + OFF_WP_ATT);
    float* Vbuf     = (float*)(ws + OFF_V);
    float* off_ws   = (float*)(ws + OFF_OFFWS);
    float* attn_ws  = (float*)(ws + OFF_ATTWS);
    float* oattn    = (float*)(ws + OFF_OATTN);

    // 1) Pack weights into WMMA B-fragment order
    pack_w_kernel<<<(256 * 256 + 255) / 256, 256, 0, stream>>>(W_value, wp_value, 256);
    pack_w_kernel<<<(256 * 256 + 255) / 256, 256, 0, stream>>>(W_out,   wp_out,   256);
    pack_w_kernel<<<(256 * 64  + 255) / 256, 256, 0, stream>>>(W_off,   wp_off,   64);
    pack_w_kernel<<<(256 * 32  + 255) / 256, 256, 0, stream>>>(W_attn,  wp_attn,  32);

    const int Mv = BS * NV;  // 80000
    const int Mq = BS * NQ;  // 80000

    // 2) Value projection: V = value @ W_value + b_value   (N=256, 2 col chunks)
    {
        int waves = Mv / 16;
        dim3 grid((waves * 32 + 255) / 256, 2);
        gemm_wmma_kernel<8, true, false><<<grid, 256, 0, stream>>>(
            value, wp_value, b_value, nullptr, Vbuf, Mv, 256);
    }
    // 3) Offset + attention-weight projections (N=64 and N=32, single chunk)
    {
        int waves = Mq / 16;
        dim3 grid((waves * 32 + 255) / 256, 1);
        gemm_wmma_kernel<4, true, false><<<grid, 256, 0, stream>>>(
            query, wp_off,  b_off,  nullptr, off_ws,  Mq, 64);
        gemm_wmma_kernel<2, true, false><<<grid, 256, 0, stream>>>(
            query, wp_attn, b_attn, nullptr, attn_ws, Mq, 32);
    }
    // 4) Softmax + bilinear deformable sampling
    {
        long long waves = (long long)BS * NQ * HEADS;          // 640000
        int blocks = (int)((waves * 32 + 255) / 256);
        msda_sample_kernel<<<blocks, 256, 0, stream>>>(Vbuf, off_ws, attn_ws, ref, oattn);
    }
    // 5) Output projection + bias + residual: out = oattn @ W_out + b_out + query
    {
        int waves = Mq / 16;
        dim3 grid((waves * 32 + 255) / 256, 2);
        gemm_wmma_kernel<8, true, true><<<grid, 256, 0, stream>>>(
            oattn, wp_out, b_out, query, out, Mq, 256);
    }
    (void)in_sizes; (void)n_in; (void)out_size; (void)ws_size;
}